// SeamContinuityLossV2_18511309046126
// MI455X (gfx1250) — compile-verified
//
#include <hip/hip_runtime.h>

// Problem constants (from the reference)
#define PP     4
#define BB     2
#define CC     3
#define HH     1024
#define WW     2048
#define NSEAM  3
#define BAND   24              // overlap rows per seam (rows 256(p+1) .. 256p+279)
#define TILE_W 256
#define LDS_R  (BAND + 2)      // +1 halo row top/bottom
#define LDS_C  (TILE_W + 2)    // +1 halo col left/right
#define EPS2   1e-6f           // EPS^2, EPS = 1e-3
#define WSEAM  1.0f
#define WGRAD  0.5f
#define WLAP   0.1f
#define DENOM  49152.0f        // 24 * 2048, same for every seam

typedef float v2f __attribute__((ext_vector_type(2)));
typedef float v8f __attribute__((ext_vector_type(8)));

__device__ __forceinline__ float robust_l1(float x) {
    return __builtin_sqrtf(x * x + EPS2);
}

// One block = one (seam, b*c, w-tile). Stages D = A - B band tile (+halo) in LDS,
// evaluates the three robust-L1 terms, reduces the wave with V_WMMA_F32_16X16X4_F32
// (B = all-ones => exact row sums), and writes one partial per block to d_ws.
__global__ __launch_bounds__(256) void seam_band_kernel(const float* __restrict__ proc,
                                                        float* __restrict__ partials)
{
    __shared__ float tile[LDS_R][LDS_C];
    __shared__ float wsum[8];

    const int tid  = threadIdx.x;
    const int wt   = blockIdx.x;   // W tile index
    const int bc   = blockIdx.y;   // b*C + c
    const int seam = blockIdx.z;   // 0..2

    const int bandStart = 256 * (seam + 1);   // first band row
    const int w0        = wt * TILE_W;

    const long long planeA = ((long long)( seam      * BB * CC + bc)) * (long long)HH * WW;
    const long long planeB = ((long long)((seam + 1) * BB * CC + bc)) * (long long)HH * WW;
    const float* __restrict__ Aimg = proc + planeA;
    const float* __restrict__ Bimg = proc + planeB;

    // Stage D tile (rows bandStart-1 .. bandStart+24) into LDS; zero-fill W halo.
    for (int i = tid; i < LDS_R * LDS_C; i += 256) {
        const int rr = i / LDS_C;
        const int cc = i - rr * LDS_C;
        const int h  = bandStart - 1 + rr;     // always in [255, 792] -> valid
        const int w  = w0 + cc - 1;
        float d = 0.0f;
        if (w >= 0 && w < WW) {
            const long long off = (long long)h * WW + w;
            d = Aimg[off] - Bimg[off];
        }
        tile[rr][cc] = d;
    }
    __syncthreads();

    // Each thread owns one column of the band.
    const int lc = tid + 1;
    float sj = 0.0f, sg = 0.0f, sl = 0.0f;
#pragma unroll 4
    for (int r = 0; r < BAND; ++r) {
        const float c  = tile[r + 1][lc];
        const float up = tile[r    ][lc];
        const float dn = tile[r + 2][lc];
        const float lf = tile[r + 1][lc - 1];
        const float rt = tile[r + 1][lc + 1];
        sj += robust_l1(c);                                  // seam jump
        sg += robust_l1(dn - c);                             // grad_h forward diff
        sl += robust_l1(up + dn + lf + rt - 4.0f * c);       // 5-point Laplacian
    }
    float partial = (WSEAM * sj + WGRAD * sg + WLAP * sl) * (1.0f / DENOM);

    // ---- Wave32 reduction via V_WMMA_F32_16X16X4_F32 (exact: multiplies by 1.0) ----
    // A (16x4): lane L<16 holds A[L, 0..1] in regs {a.x, a.y}; lane L>=16 holds A[L-16, 2..3].
    // With a = {partial, 0}: row m sums to partial[lane m] + partial[lane m+16].
    // B = all ones (4x16) => D[m][n] = rowsum(m) for every n.
    v2f a;     a.x = partial; a.y = 0.0f;
    v2f bones; bones.x = 1.0f; bones.y = 1.0f;
    v8f cz = {0.f, 0.f, 0.f, 0.f, 0.f, 0.f, 0.f, 0.f};
    v8f dm = __builtin_amdgcn_wmma_f32_16x16x4_f32(
        /*neg_a=*/false, a, /*neg_b=*/false, bones,
        /*c_mod=*/(short)0, cz, /*reuse_a=*/false, /*reuse_b=*/false);

    // D VGPR j at lane l holds D[M = j + 8*(l/16)][N = l%16]:
    // summing the 8 regs gives rows 0..7 (lanes 0-15) or rows 8..15 (lanes 16-31).
    float s = dm[0] + dm[1] + dm[2] + dm[3] + dm[4] + dm[5] + dm[6] + dm[7];
    const float waveSum = __shfl(s, 0, 32) + __shfl(s, 16, 32);

    const int wave = tid >> 5;
    const int lane = tid & 31;
    if (lane == 0) wsum[wave] = waveSum;
    __syncthreads();

    if (tid == 0) {
        float bsum = 0.0f;
#pragma unroll
        for (int i = 0; i < 8; ++i) bsum += wsum[i];
        const int blockId = (blockIdx.z * gridDim.y + blockIdx.y) * gridDim.x + blockIdx.x;
        partials[blockId] = bsum;
    }
}

// Deterministic final reduction of the 144 per-block partials (no float atomics,
// so every graph replay produces bit-identical output).
__global__ __launch_bounds__(256) void seam_reduce_kernel(const float* __restrict__ partials,
                                                          float* __restrict__ out, int n)
{
    __shared__ float sh[256];
    const int tid = threadIdx.x;
    float v = 0.0f;
    for (int i = tid; i < n; i += 256) v += partials[i];
    sh[tid] = v;
    __syncthreads();
    for (int stride = 128; stride > 0; stride >>= 1) {
        if (tid < stride) sh[tid] += sh[tid + stride];
        __syncthreads();
    }
    if (tid == 0) out[0] = sh[0];
}

extern "C" void kernel_launch(void* const* d_in, const int* in_sizes, int n_in,
                              void* d_out, int out_size, void* d_ws, size_t ws_size,
                              hipStream_t stream)
{
    const float* proc = (const float*)d_in[0];
    // d_in[1] is the mask; the band structure (rows 256(p+1)..256p+279, 24 rows,
    // denom = 24*2048) is fixed by the reference's SEGMENTS, so it's folded in
    // statically — we never touch the other ~980 rows per seam (~50x less HBM traffic).
    float* out      = (float*)d_out;
    float* partials = (float*)d_ws;   // 144 floats of scratch

    dim3 grid(WW / TILE_W, BB * CC, NSEAM);   // (8, 6, 3) = 144 blocks
    seam_band_kernel<<<grid, 256, 0, stream>>>(proc, partials);

    const int nPartials = (WW / TILE_W) * BB * CC * NSEAM;
    seam_reduce_kernel<<<1, 256, 0, stream>>>(partials, out, nPartials);
}